// FlowAttention_39032662786432
// MI455X (gfx1250) — compile-verified
//
#include <hip/hip_runtime.h>
#include <hip/hip_bf16.h>
#include <stddef.h>
#include <stdint.h>

// FlowAttention for MI455X (gfx1250, wave32).
// GEMMs on V_WMMA_F32_16X16X4_F32; A-tile staging via CDNA5 async-to-LDS
// (GLOBAL_LOAD_ASYNC_TO_LDS_B128, GVS mode, S_WAIT_ASYNCCNT); B-tile stored
// transposed in LDS so all WMMA fragments are single ds_load_b64 ops.

typedef __attribute__((ext_vector_type(2))) float v2f;
typedef __attribute__((ext_vector_type(4))) float f4;
typedef __attribute__((ext_vector_type(8))) float v8f;

#define EPSV 1e-6f

// Async DMA global->LDS, GVS mode: global address is the SGPR base plus a 32-bit VGPR byte offset,
// LDS address comes from the VDST VGPR. Per-thread VGPR offsets are loop-invariant;
// only the uniform SGPR base advances per k-tile. Tracked by ASYNCcnt.
__device__ __forceinline__ void async_copy_b128_gvs(uint32_t lds_off,
                                                    uint32_t voff,
                                                    const float* sbase) {
    asm volatile("global_load_async_to_lds_b128 %0, %1, %2"
                 :: "v"(lds_off), "v"(voff), "s"(sbase)
                 : "memory");
}
__device__ __forceinline__ void wait_async_all() {
    asm volatile("s_wait_asynccnt 0x0" ::: "memory");
}

// ---------------------------------------------------------------------------
// fp32 GEMM: C[M,N] = A[M,K] @ B[K,N] (+ bias[N] if bias != nullptr)
// Block tile 256x128, K-tile 32. 256 threads = 8 waves; each wave owns a
// 64x64 subtile = 4x4 WMMA tiles (16 wmma per 4-deep k-step, 8 b64 frag loads).
// A LDS stride 36: 16B-aligned async writes, 8B-aligned frags, conflict-free
// 16-lane column reads. B kept transposed (Bt[c][k], stride 38) so B frags
// are contiguous b64 as well. M % 256 == 0, N % 128 == 0, K % 32 == 0.
// ---------------------------------------------------------------------------
#define GEMM_BM 256
#define GEMM_BN 128
#define GEMM_BK 32
#define A_STRIDE 36
#define BT_STRIDE 38

__global__ __launch_bounds__(256) void gemm_f32_wmma(
    const float* __restrict__ A, const float* __restrict__ B,
    const float* __restrict__ bias, float* __restrict__ C,
    int M, int N, int K)
{
    __shared__ float As[GEMM_BM][A_STRIDE];    // 36 KB
    __shared__ float Bt[GEMM_BN][BT_STRIDE];   // 19 KB, transposed B tile

    const int tid  = threadIdx.x;
    const int lane = tid & 31;
    const int wave = tid >> 5;
    const int l16  = lane & 15;
    const int lh   = lane >> 4;       // 0/1: K-pair half & M+8 half
    const int wr   = wave & 3;        // 4 row groups of 64
    const int wc   = wave >> 2;       // 2 col groups of 64

    const int m0 = blockIdx.y * GEMM_BM;
    const int n0 = blockIdx.x * GEMM_BN;

    // Loop-invariant async staging offsets for the A tile (bytes).
    const int arow = tid >> 3;               // 0..31
    const int akq  = (tid & 7) << 2;         // 0,4,...,28
    const uint32_t a_voff = (uint32_t)((arow * K + akq) * 4);
    const uint32_t a_loff = (uint32_t)(uintptr_t)&As[arow][akq];

    // B staging indices (through VGPRs, stored transposed).
    const int bkrow = tid >> 5;              // 0..7
    const int bcq   = (tid & 31) << 2;       // 0..124

    v8f zero = {};
    v8f acc[4][4];
#pragma unroll
    for (int i = 0; i < 4; ++i)
#pragma unroll
        for (int j = 0; j < 4; ++j) acc[i][j] = zero;

    for (int k0 = 0; k0 < K; k0 += GEMM_BK) {
        // A tile 256x32 via async DMA: 8 chunks/thread, SGPR base per tile.
        const float* sA = A + (size_t)m0 * K + k0;
#pragma unroll
        for (int i = 0; i < 8; ++i)
            async_copy_b128_gvs(a_loff + (uint32_t)(i * 32 * A_STRIDE * 4),
                                a_voff + (uint32_t)(i * 32 * K * 4), sA);

        // B tile 32x128 -> Bt[c][k]: 4 float4/thread, strided scalar stores.
#pragma unroll
        for (int i = 0; i < 4; ++i) {
            int krow = bkrow + i * 8;
            f4 bv = *(const f4*)(B + (size_t)(k0 + krow) * N + n0 + bcq);
            Bt[bcq + 0][krow] = bv.x;
            Bt[bcq + 1][krow] = bv.y;
            Bt[bcq + 2][krow] = bv.z;
            Bt[bcq + 3][krow] = bv.w;
        }
        wait_async_all();
        __syncthreads();

#pragma unroll
        for (int kk = 0; kk < GEMM_BK; kk += 4) {
            const int kkk = kk + 2 * lh;
            v2f afrag[4], bfrag[4];
#pragma unroll
            for (int mt = 0; mt < 4; ++mt) {
                int r = wr * 64 + mt * 16 + l16;
                afrag[mt] = *(const v2f*)&As[r][kkk];   // ds_load_b64
            }
#pragma unroll
            for (int nt = 0; nt < 4; ++nt) {
                int c = wc * 64 + nt * 16 + l16;
                bfrag[nt] = *(const v2f*)&Bt[c][kkk];   // ds_load_b64
            }
#pragma unroll
            for (int mt = 0; mt < 4; ++mt)
#pragma unroll
                for (int nt = 0; nt < 4; ++nt)
                    acc[mt][nt] = __builtin_amdgcn_wmma_f32_16x16x4_f32(
                        false, afrag[mt], false, bfrag[nt],
                        (short)0, acc[mt][nt], false, false);
        }
        __syncthreads();
    }

#pragma unroll
    for (int mt = 0; mt < 4; ++mt)
#pragma unroll
        for (int nt = 0; nt < 4; ++nt)
#pragma unroll
            for (int r = 0; r < 8; ++r) {
                int row = m0 + wr * 64 + mt * 16 + r + 8 * lh;
                int col = n0 + wc * 64 + nt * 16 + l16;
                float v = acc[mt][nt][r];
                if (bias) v += bias[col];
                C[(size_t)row * N + col] = v;
            }
}

// ---------------------------------------------------------------------------
// Per (b,h): kv[d,d] = sum_n relu(k[n,:])^T v[n,:]  and  ksum[d] = sum relu(k)
// One workgroup per (b,h). K/V staged 64(n) x 64(d) through LDS; kv via WMMA
// (A = k^T read column-wise from LDS), ksum via VALU partials.
// ---------------------------------------------------------------------------
__global__ __launch_bounds__(256) void kv_ksum_kernel(
    const float* __restrict__ qkv, float* __restrict__ kvout,
    float* __restrict__ ksum, int N)
{
    __shared__ float ks[64][64];
    __shared__ float vs[64][64];
    __shared__ float red[4][64];

    const int bh = blockIdx.x;           // 0..63
    const int b  = bh >> 4;
    const int h  = bh & 15;

    const int tid  = threadIdx.x;
    const int lane = tid & 31;
    const int wave = tid >> 5;
    const int l16  = lane & 15;
    const int lh   = lane >> 4;
    const int mt   = wave & 3;           // output tile row  (of 4)
    const int ntb  = (wave >> 2) * 2;    // output tile col base (2 tiles/wave)

    const int dim = tid & 63;
    const int seg = tid >> 6;
    float psum = 0.f;

    v8f zero = {};
    v8f acc[2] = {zero, zero};

    const float* kbase = qkv + (size_t)b * N * 3072 + 1024 + h * 64;
    const float* vbase = qkv + (size_t)b * N * 3072 + 2048 + h * 64;

    for (int n0 = 0; n0 < N; n0 += 64) {
        // Stage 64x64 K (ReLU fused) and V tiles: 1024 float4 each, 4/thread
#pragma unroll
        for (int i = 0; i < 4; ++i) {
            int idx = tid + i * 256;
            int row = idx >> 4;
            int cq  = (idx & 15) << 2;
            f4 kv4 = *(const f4*)(kbase + (size_t)(n0 + row) * 3072 + cq);
            ks[row][cq + 0] = fmaxf(kv4.x, 0.f);
            ks[row][cq + 1] = fmaxf(kv4.y, 0.f);
            ks[row][cq + 2] = fmaxf(kv4.z, 0.f);
            ks[row][cq + 3] = fmaxf(kv4.w, 0.f);
            *(f4*)&vs[row][cq] = *(const f4*)(vbase + (size_t)(n0 + row) * 3072 + cq);
        }
        __syncthreads();

        // ksum partials: 4 threads per dim, 16 rows each
#pragma unroll
        for (int r = 0; r < 16; ++r) psum += ks[seg * 16 + r][dim];

        // kv += k^T v : K-dim = n (64), A(m,n') = ks[n'][m], B(n',c) = vs[n'][c]
#pragma unroll
        for (int kk = 0; kk < 64; kk += 4) {
            const int kkk = kk + 2 * lh;
            v2f a;
            a.x = ks[kkk][mt * 16 + l16];
            a.y = ks[kkk + 1][mt * 16 + l16];
#pragma unroll
            for (int j = 0; j < 2; ++j) {
                int c = (ntb + j) * 16 + l16;
                v2f bf;
                bf.x = vs[kkk][c];
                bf.y = vs[kkk + 1][c];
                acc[j] = __builtin_amdgcn_wmma_f32_16x16x4_f32(
                    false, a, false, bf, (short)0, acc[j], false, false);
            }
        }
        __syncthreads();
    }

    // write kv tile
#pragma unroll
    for (int j = 0; j < 2; ++j)
#pragma unroll
        for (int r = 0; r < 8; ++r) {
            int row = mt * 16 + r + 8 * lh;
            int col = (ntb + j) * 16 + l16;
            kvout[(size_t)bh * 4096 + row * 64 + col] = acc[j][r];
        }

    // reduce ksum
    red[seg][dim] = psum;
    __syncthreads();
    if (tid < 64)
        ksum[bh * 64 + tid] =
            red[0][tid] + red[1][tid] + red[2][tid] + red[3][tid] + EPSV;
}

// ---------------------------------------------------------------------------
// Per (b,h, 128-row block): y = (relu(q) @ kv) * z, z = 1/(relu(q).ksum + eps)
// Written into y[B,N,D] layout so the output projection is a plain GEMM.
// ---------------------------------------------------------------------------
__global__ __launch_bounds__(256) void attn_out_kernel(
    const float* __restrict__ qkv, const float* __restrict__ kvin,
    const float* __restrict__ ksum, float* __restrict__ y, int N)
{
    __shared__ float qs[128][64];
    __shared__ float kvs[64][64];
    __shared__ float ksh[64];
    __shared__ float zs[128];

    const int bh = blockIdx.x;           // 0..63
    const int b  = bh >> 4;
    const int h  = bh & 15;
    const int n0 = blockIdx.y * 128;

    const int tid  = threadIdx.x;
    const int lane = tid & 31;
    const int wave = tid >> 5;
    const int l16  = lane & 15;
    const int lh   = lane >> 4;
    const int mb   = (wave >> 1) * 32;   // wave row base (4 groups of 32)
    const int nb   = (wave & 1) * 32;    // wave col base (2 groups of 32)

    const float* qbase = qkv + (size_t)b * N * 3072 + h * 64;

    // stage kv (64x64) and ksum
#pragma unroll
    for (int i = 0; i < 4; ++i) {
        int idx = tid + i * 256;
        *(f4*)(&kvs[0][0] + idx * 4) = *(const f4*)(kvin + (size_t)bh * 4096 + idx * 4);
    }
    if (tid < 16)
        *(f4*)&ksh[tid * 4] = *(const f4*)(ksum + bh * 64 + tid * 4);

    // stage q tile 128x64 with ReLU (2048 float4, 8/thread)
#pragma unroll
    for (int i = 0; i < 8; ++i) {
        int idx = tid + i * 256;
        int row = idx >> 4;
        int cq  = (idx & 15) << 2;
        f4 qv = *(const f4*)(qbase + (size_t)(n0 + row) * 3072 + cq);
        qs[row][cq + 0] = fmaxf(qv.x, 0.f);
        qs[row][cq + 1] = fmaxf(qv.y, 0.f);
        qs[row][cq + 2] = fmaxf(qv.z, 0.f);
        qs[row][cq + 3] = fmaxf(qv.w, 0.f);
    }
    __syncthreads();

    // z per row
    if (tid < 128) {
        float s = 0.f;
#pragma unroll
        for (int c = 0; c < 64; ++c) s += qs[tid][c] * ksh[c];
        zs[tid] = 1.f / (s + EPSV);
    }
    __syncthreads();

    // 128x64 = 8x4 tiles; each wave 2x2 tiles, K = 64
    v8f zero = {};
    v8f acc[2][2] = {{zero, zero}, {zero, zero}};
#pragma unroll
    for (int kk = 0; kk < 64; kk += 4) {
        const int kkk = kk + 2 * lh;
        v2f a[2], bf[2];
#pragma unroll
        for (int i = 0; i < 2; ++i) {
            int r = mb + i * 16 + l16;
            a[i].x = qs[r][kkk];
            a[i].y = qs[r][kkk + 1];
            int c = nb + i * 16 + l16;
            bf[i].x = kvs[kkk][c];
            bf[i].y = kvs[kkk + 1][c];
        }
#pragma unroll
        for (int i = 0; i < 2; ++i)
#pragma unroll
            for (int j = 0; j < 2; ++j)
                acc[i][j] = __builtin_amdgcn_wmma_f32_16x16x4_f32(
                    false, a[i], false, bf[j], (short)0, acc[i][j], false, false);
    }

#pragma unroll
    for (int i = 0; i < 2; ++i)
#pragma unroll
        for (int j = 0; j < 2; ++j)
#pragma unroll
            for (int r = 0; r < 8; ++r) {
                int rowl = mb + i * 16 + r + 8 * lh;
                int col  = nb + j * 16 + l16;
                y[((size_t)b * N + n0 + rowl) * 1024 + h * 64 + col] =
                    acc[i][j][r] * zs[rowl];
            }
}

// ---------------------------------------------------------------------------
extern "C" void kernel_launch(void* const* d_in, const int* in_sizes, int n_in,
                              void* d_out, int out_size, void* d_ws, size_t ws_size,
                              hipStream_t stream) {
    const float* x    = (const float*)d_in[0];   // [4,4096,1024]
    const float* Wqkv = (const float*)d_in[1];   // [1024,3072]
    const float* Wout = (const float*)d_in[2];   // [1024,1024]
    const float* bout = (const float*)d_in[3];   // [1024]
    float* out = (float*)d_out;                  // [4,4096,1024]

    const int Bb = 4, N = 4096, D = 1024;
    const int M  = Bb * N;                       // 16384

    float* qkv  = (float*)d_ws;                  // [M, 3D]   201.3 MB
    float* yb   = qkv + (size_t)M * 3 * D;       // [M, D]     67.1 MB
    float* kvws = yb + (size_t)M * D;            // [64,64,64]  1.0 MB
    float* ksws = kvws + (size_t)64 * 64 * 64;   // [64,64]

    // 1) qkv = x @ Wqkv                      (103 GFLOP, WMMA f32)
    gemm_f32_wmma<<<dim3(3 * D / GEMM_BN, M / GEMM_BM), 256, 0, stream>>>(
        x, Wqkv, nullptr, qkv, M, 3 * D, D);

    // 2) kv = relu(k)^T v, ksum = sum relu(k) + eps   (per b,h)
    kv_ksum_kernel<<<dim3(64), 256, 0, stream>>>(qkv, kvws, ksws, N);

    // 3) y = (relu(q) @ kv) * z
    attn_out_kernel<<<dim3(64, N / 128), 256, 0, stream>>>(qkv, kvws, ksws, yb, N);

    // 4) out = y @ Wout + bout               (34 GFLOP, WMMA f32)
    gemm_f32_wmma<<<dim3(D / GEMM_BN, M / GEMM_BM), 256, 0, stream>>>(
        yb, Wout, bout, out, M, D, D);
}